// Bil_layer_84284438216853
// MI455X (gfx1250) — compile-verified
//
#include <hip/hip_runtime.h>
#include <hip/hip_bf16.h>
#include <stdint.h>

// ---------------------------------------------------------------------------
// Bil_layer fused pipeline on (16,3,512,512) fp32:
//   bilateral3(sigma=0.1) -> bilateral3(sigma=0.01) -> median3 -> median3
// All 4 radius-1 stages fused in ONE kernel: 32x32 output tile, 40x40x3 input
// halo staged to LDS via gfx1250 async global->LDS loads (ASYNCcnt), stage
// intermediates ping-pong in LDS. HBM traffic = 1 read + 1 write of the
// tensor (~100 MB -> ~4.3 us floor at 23.3 TB/s) vs 4 passes (~400 MB).
// Reflect padding handled once at radius 4; stencil mirror-symmetry makes
// this exactly equivalent to per-stage reflect padding.
// No WMMA: weights are data-dependent, median is a sorting network.
// ---------------------------------------------------------------------------

#define TW 32
#define TH 32
#define NT 256
#define RAD 4                 // total halo radius (4 radius-1 stages)
#define AW  (TW + 2 * RAD)    // 40 : input tile width
#define AN  (AW * AW)         // 1600
#define S1W (AW - 2)          // 38 : bilateral-1 output
#define S2W (AW - 4)          // 36 : bilateral-2 output
#define S3W (AW - 6)          // 34 : median-1 output

// ---- gfx1250 async global->LDS (VFLAT async ops, tracked by ASYNCcnt) -----
// LDS byte address = low 32 bits of the flat shared-aperture pointer
// (see ISA section 10.2).
__device__ __forceinline__ void async_ld_f32(const float* g, const float* lds) {
  uint32_t lds_off = (uint32_t)(uintptr_t)lds;
  asm volatile("global_load_async_to_lds_b32 %0, %1, off"
               :: "v"(lds_off), "v"(g) : "memory");
}

__device__ __forceinline__ void async_ld_f32x4(const float* g, const float* lds) {
  uint32_t lds_off = (uint32_t)(uintptr_t)lds;
  asm volatile("global_load_async_to_lds_b128 %0, %1, off"
               :: "v"(lds_off), "v"(g) : "memory");
}

__device__ __forceinline__ void wait_async0() {
#if __has_builtin(__builtin_amdgcn_s_wait_asynccnt)
  __builtin_amdgcn_s_wait_asynccnt(0);
#else
  asm volatile("s_wait_asynccnt 0" ::: "memory");
#endif
}

// jnp.pad 'reflect' (no edge duplication): -i -> i, N-1+k -> N-1-k.
__device__ __forceinline__ int reflect_idx(int i, int n) {
  i = (i < 0) ? -i : i;
  if (i >= n) i = 2 * n - 2 - i;
  return i;
}

// ---- bilateral stage: 3-channel coupled, l1 color distance ----------------
template <int SRCW, int DSTW>
__device__ __forceinline__ void bilateral_stage(const float* __restrict__ src,
                                                float* __restrict__ dst,
                                                float scf, const float* w1d,
                                                int tid) {
  const int sp = SRCW * SRCW;   // src channel plane
  const int dp = DSTW * DSTW;   // dst channel plane
  for (int i = tid; i < dp; i += NT) {
    const int y = i / DSTW;
    const int x = i - y * DSTW;
    const float* s = src + y * SRCW + x;  // 3x3 window top-left
    const float c0 = s[SRCW + 1];
    const float c1 = s[sp + SRCW + 1];
    const float c2 = s[2 * sp + SRCW + 1];

    float wsum = 0.0f, a0 = 0.0f, a1 = 0.0f, a2 = 0.0f;
#pragma unroll
    for (int dy = 0; dy < 3; ++dy) {
#pragma unroll
      for (int dx = 0; dx < 3; ++dx) {
        const int o = dy * SRCW + dx;
        const float n0 = s[o];
        const float n1 = s[sp + o];
        const float n2 = s[2 * sp + o];
        const float cd = fabsf(n0 - c0) + fabsf(n1 - c1) + fabsf(n2 - c2);
        const float w  = __expf(scf * cd * cd) * (w1d[dy] * w1d[dx]);
        wsum += w;
        a0 = fmaf(n0, w, a0);
        a1 = fmaf(n1, w, a1);
        a2 = fmaf(n2, w, a2);
      }
    }
    const float inv = 1.0f / wsum;
    dst[i]           = a0 * inv;
    dst[dp + i]      = a1 * inv;
    dst[2 * dp + i]  = a2 * inv;
  }
}

// ---- median-of-9: 19-exchange min/max network -----------------------------
__device__ __forceinline__ void mm2(float& a, float& b) {
  const float t = fminf(a, b);
  b = fmaxf(a, b);
  a = t;
}

__device__ __forceinline__ float median9(float p0, float p1, float p2,
                                         float p3, float p4, float p5,
                                         float p6, float p7, float p8) {
  mm2(p1, p2); mm2(p4, p5); mm2(p7, p8);
  mm2(p0, p1); mm2(p3, p4); mm2(p6, p7);
  mm2(p1, p2); mm2(p4, p5); mm2(p7, p8);
  mm2(p0, p3); mm2(p5, p8); mm2(p4, p7);
  mm2(p3, p6); mm2(p1, p4); mm2(p2, p5);
  mm2(p4, p7); mm2(p4, p2); mm2(p6, p4);
  mm2(p4, p2);
  return p4;
}

template <int SRCW, int DSTW>
__device__ __forceinline__ void median_stage(const float* __restrict__ src,
                                             float* __restrict__ dst, int tid) {
  const int sp = SRCW * SRCW;
  const int dp = DSTW * DSTW;
  for (int i = tid; i < 3 * dp; i += NT) {
    const int c = i / dp;
    const int r = i - c * dp;
    const int y = r / DSTW;
    const int x = r - y * DSTW;
    const float* s = src + c * sp + y * SRCW + x;
    dst[i] = median9(s[0],            s[1],            s[2],
                     s[SRCW],         s[SRCW + 1],     s[SRCW + 2],
                     s[2 * SRCW],     s[2 * SRCW + 1], s[2 * SRCW + 2]);
  }
}

// ---------------------------------------------------------------------------
__global__ __launch_bounds__(NT) void bil_layer_fused_kernel(
    const float* __restrict__ in, float* __restrict__ out,
    int H, int W, float sc1, float sc2) {
  // 16B alignment required: B128 async-to-LDS destinations.
  __shared__ __align__(16) float bufA[3 * AN];          // 40x40x3 in, 36x36x3 s2
  __shared__ __align__(16) float bufB[3 * S1W * S1W];   // 38x38x3 s1, 34x34x3 s3

  const int tx0 = blockIdx.x * TW;
  const int ty0 = blockIdx.y * TH;
  const int b   = blockIdx.z;
  const int tid = threadIdx.x;
  const size_t plane = (size_t)H * W;
  const float* base = in + (size_t)b * 3 * plane;

  // ---- stage 0: async-stage 40x40x3 halo into LDS -------------------------
  const bool interior = (tx0 >= RAD) && (ty0 >= RAD) &&
                        (tx0 + TW + RAD <= W) && (ty0 + TH + RAD <= H);
  if (interior) {
    // contiguous rows; (tx0-4) is a multiple of 4 floats -> 16B aligned B128
    const float* org = base + (size_t)(ty0 - RAD) * W + (tx0 - RAD);
    const int vecs_per_row = AW / 4;                 // 10
    const int nvec = 3 * AW * vecs_per_row;          // 1200
    for (int v = tid; v < nvec; v += NT) {
      const int c    = v / (AW * vecs_per_row);
      const int r    = v - c * (AW * vecs_per_row);
      const int row  = r / vecs_per_row;
      const int col4 = (r - row * vecs_per_row) * 4;
      async_ld_f32x4(org + c * plane + (size_t)row * W + col4,
                     &bufA[c * AN + row * AW + col4]);
    }
  } else {
    // border tile: per-lane reflect-mapped scalar async loads
    for (int i = tid; i < 3 * AN; i += NT) {
      const int c  = i / AN;
      const int r  = i - c * AN;
      const int yy = r / AW;
      const int xx = r - yy * AW;
      const int gy = reflect_idx(ty0 - RAD + yy, H);
      const int gx = reflect_idx(tx0 - RAD + xx, W);
      async_ld_f32(base + c * plane + (size_t)gy * W + gx, &bufA[i]);
    }
  }
  wait_async0();
  __syncthreads();

  // Separable spatial Gaussian, sigma_space = 10, kernel 3, normalized.
  const float g1  = __expf(-0.005f);                 // exp(-0.5*(1/10)^2)
  const float nrm = 1.0f / (2.0f * g1 + 1.0f);
  const float w1d[3] = {g1 * nrm, nrm, g1 * nrm};

  // ---- stage 1: bilateral sigma=0.1  : A(40) -> B(38) ---------------------
  bilateral_stage<AW, S1W>(bufA, bufB, sc1, w1d, tid);
  __syncthreads();

  // ---- stage 2: bilateral sigma=0.01 : B(38) -> A(36) ---------------------
  bilateral_stage<S1W, S2W>(bufB, bufA, sc2, w1d, tid);
  __syncthreads();

  // ---- stage 3: median : A(36) -> B(34) -----------------------------------
  median_stage<S2W, S3W>(bufA, bufB, tid);
  __syncthreads();

  // ---- stage 4: median : B(34) -> global, 4 pixels/thread, B128 stores ----
  // 3*32*32 outputs as 768 float4 groups; (tx0 + 4k) is 16B aligned.
  const int sp3 = S3W * S3W;
  for (int i = tid; i < 3 * TW * TH / 4; i += NT) {     // 768 vec4 items
    const int c  = i >> 8;                              // / 256
    const int r  = i & 255;
    const int y  = r >> 3;                              // 8 vec4 per row
    const int x4 = (r & 7) * 4;
    const float* s = bufB + c * sp3 + y * S3W + x4;     // window top-left col

    // 6 columns x 3 rows shared by the 4 adjacent medians
    float r0[6], r1[6], r2[6];
#pragma unroll
    for (int k = 0; k < 6; ++k) {
      r0[k] = s[k];
      r1[k] = s[S3W + k];
      r2[k] = s[2 * S3W + k];
    }

    float4 m;
    m.x = median9(r0[0], r0[1], r0[2], r1[0], r1[1], r1[2], r2[0], r2[1], r2[2]);
    m.y = median9(r0[1], r0[2], r0[3], r1[1], r1[2], r1[3], r2[1], r2[2], r2[3]);
    m.z = median9(r0[2], r0[3], r0[4], r1[2], r1[3], r1[4], r2[2], r2[3], r2[4]);
    m.w = median9(r0[3], r0[4], r0[5], r1[3], r1[4], r1[5], r2[3], r2[4], r2[5]);

    float4* dst = (float4*)(out + ((size_t)(b * 3 + c) * H + (ty0 + y)) * W
                                + (tx0 + x4));
    *dst = m;
  }
}

// ---------------------------------------------------------------------------
extern "C" void kernel_launch(void* const* d_in, const int* in_sizes, int n_in,
                              void* d_out, int out_size, void* d_ws, size_t ws_size,
                              hipStream_t stream) {
  (void)in_sizes; (void)n_in; (void)out_size; (void)d_ws; (void)ws_size;
  const int B = 16, H = 512, W = 512;

  const float* x   = (const float*)d_in[0];
  float*       out = (float*)d_out;

  const dim3 blk(NT);
  const dim3 grid(W / TW, H / TH, B);

  const float sc1 = -0.5f / (0.1f * 0.1f);    // first bilateral pass
  const float sc2 = -0.5f / (0.01f * 0.01f);  // second bilateral pass

  bil_layer_fused_kernel<<<grid, blk, 0, stream>>>(x, out, H, W, sc1, sc2);
}